// PathwaySupConLoss_54709293416904
// MI455X (gfx1250) — compile-verified
//
#include <hip/hip_runtime.h>
#include <math.h>

#define O_ 3
#define P_ 128
#define G_ 64
#define N_ 20000
#define H_ 512
#define D_ 128
#define M_ (P_ * O_)        // 384
#define INV_T 10.0f         // 1 / TEMPERATURE
#define EPS_ 1e-12f

typedef __attribute__((ext_vector_type(2))) float v2f;
typedef __attribute__((ext_vector_type(8))) float v8f;

static __device__ __forceinline__ v8f wmma_f32(v2f a, v2f b, v8f c) {
  return __builtin_amdgcn_wmma_f32_16x16x4_f32(false, a, false, b,
                                               (short)0, c, false, false);
}

// ---------------------------------------------------------------------------
// Kernel 1: protos[o,p,:] = mean_g emb[o, idx[o,p,g], :]
// ---------------------------------------------------------------------------
__global__ void gather_mean_kernel(const float* __restrict__ emb,
                                   const int* __restrict__ idx,
                                   float* __restrict__ protos) {
  const int op = blockIdx.x;            // 0..383
  const int o = op / P_;
  const int p = op % P_;
  const int* ip = idx + (o * P_ + p) * G_;
  const float* eo = emb + (size_t)o * N_ * H_;
  const int c0 = threadIdx.x;           // columns c0 and c0+256
  float acc0 = 0.0f, acc1 = 0.0f;
  for (int g = 0; g < G_; ++g) {
    const float* row = eo + (size_t)ip[g] * H_;
    acc0 += row[c0];
    acc1 += row[c0 + 256];
  }
  float* out = protos + (size_t)(o * P_ + p) * H_;
  out[c0]       = acc0 * (1.0f / G_);
  out[c0 + 256] = acc1 * (1.0f / G_);
}

// ---------------------------------------------------------------------------
// Kernel 2: batched transpose  in: [O][R][C]  ->  out: [O][C][R]
// Coalesced reads; used once for W1 (R=H,C=H) and W2 (R=H,C=D).
// ---------------------------------------------------------------------------
__global__ void transpose_kernel(const float* __restrict__ in,
                                 float* __restrict__ out, int R, int C) {
  const int tid = blockIdx.x * blockDim.x + threadIdx.x;
  const int total = O_ * R * C;
  if (tid >= total) return;
  const int o  = tid / (R * C);
  const int rc = tid % (R * C);
  const int r  = rc / C;
  const int c  = rc % C;
  out[(size_t)o * R * C + (size_t)c * R + r] = in[tid];
}

// ---------------------------------------------------------------------------
// Kernel 3: h[o] = relu(protos[o] (128x512) @ W1[o] (512x512) + b1[o])
// Each wave: 64x16 output strip = 4 accumulators sharing one B operand.
// B comes from transposed W1t[o][n][k] -> single b64 load per k-step.
// Software-pipelined k-loop. 192 waves -> 24 blocks * 8 waves.
// ---------------------------------------------------------------------------
__global__ void gemm1_relu_kernel(const float* __restrict__ protos,
                                  const float* __restrict__ W1t,
                                  const float* __restrict__ b1,
                                  float* __restrict__ hbuf) {
  const int lane = threadIdx.x & 31;
  const int wv   = threadIdx.x >> 5;
  const int w    = blockIdx.x * 8 + wv;     // 0..191
  const int o    = w / 64;                  // 64 waves per omics
  const int rem  = w % 64;
  const int tm   = (rem / 32) * 64;         // 2 strips of 64 rows
  const int tn   = (rem % 32) * 16;         // 32 tiles along N=512
  const float* A  = protos + (size_t)o * P_ * H_;   // [m][k]
  const float* Bt = W1t    + (size_t)o * H_ * H_;   // [n][k]
  const int half = lane >> 4;               // K pair: 0->{0,1}, 1->{2,3}
  const int l15  = lane & 15;

  const float* ar0 = A  + (size_t)(tm +  0 + l15) * H_;
  const float* ar1 = A  + (size_t)(tm + 16 + l15) * H_;
  const float* ar2 = A  + (size_t)(tm + 32 + l15) * H_;
  const float* ar3 = A  + (size_t)(tm + 48 + l15) * H_;
  const float* br  = Bt + (size_t)(tn + l15) * H_;

  v8f c0 = {}, c1 = {}, c2 = {}, c3 = {};
  int klo = half * 2;
  v2f a0 = *(const v2f*)(ar0 + klo);
  v2f a1 = *(const v2f*)(ar1 + klo);
  v2f a2 = *(const v2f*)(ar2 + klo);
  v2f a3 = *(const v2f*)(ar3 + klo);
  v2f b  = *(const v2f*)(br  + klo);
  for (int k = 0; k < H_ - 4; k += 4) {
    const int kn = k + 4 + half * 2;
    v2f na0 = *(const v2f*)(ar0 + kn);
    v2f na1 = *(const v2f*)(ar1 + kn);
    v2f na2 = *(const v2f*)(ar2 + kn);
    v2f na3 = *(const v2f*)(ar3 + kn);
    v2f nb  = *(const v2f*)(br  + kn);
    c0 = wmma_f32(a0, b, c0);
    c1 = wmma_f32(a1, b, c1);
    c2 = wmma_f32(a2, b, c2);
    c3 = wmma_f32(a3, b, c3);
    a0 = na0; a1 = na1; a2 = na2; a3 = na3; b = nb;
  }
  c0 = wmma_f32(a0, b, c0);
  c1 = wmma_f32(a1, b, c1);
  c2 = wmma_f32(a2, b, c2);
  c3 = wmma_f32(a3, b, c3);

  const int col = tn + l15;
  const float bias = b1[o * H_ + col];
  float* Hb = hbuf + (size_t)o * P_ * H_;
  v8f acc[4] = {c0, c1, c2, c3};
  for (int i = 0; i < 4; ++i)
    for (int v = 0; v < 8; ++v) {
      const int row = tm + 16 * i + v + half * 8;   // C: M = v + 8*(lane>=16)
      Hb[(size_t)row * H_ + col] = fmaxf(acc[i][v] + bias, 0.0f);
    }
}

// ---------------------------------------------------------------------------
// Kernel 4: z[o] = h[o] (128x512) @ W2[o] (512x128) + b2[o]
// Same 64x16-per-wave scheme with W2t[o][n][k]. Rows stored at m = p*O + o.
// 48 waves -> 6 blocks * 8 waves.
// ---------------------------------------------------------------------------
__global__ void gemm2_kernel(const float* __restrict__ hbuf,
                             const float* __restrict__ W2t,
                             const float* __restrict__ b2,
                             float* __restrict__ Z) {
  const int lane = threadIdx.x & 31;
  const int wv   = threadIdx.x >> 5;
  const int w    = blockIdx.x * 8 + wv;     // 0..47
  const int o    = w / 16;                  // 16 waves per omics
  const int rem  = w % 16;
  const int tm   = (rem / 8) * 64;          // 2 strips of 64 rows
  const int tn   = (rem % 8) * 16;          // 8 tiles along N=128
  const float* A  = hbuf + (size_t)o * P_ * H_;   // [m][k]
  const float* Bt = W2t  + (size_t)o * D_ * H_;   // [n][k]
  const int half = lane >> 4;
  const int l15  = lane & 15;

  const float* ar0 = A  + (size_t)(tm +  0 + l15) * H_;
  const float* ar1 = A  + (size_t)(tm + 16 + l15) * H_;
  const float* ar2 = A  + (size_t)(tm + 32 + l15) * H_;
  const float* ar3 = A  + (size_t)(tm + 48 + l15) * H_;
  const float* br  = Bt + (size_t)(tn + l15) * H_;

  v8f c0 = {}, c1 = {}, c2 = {}, c3 = {};
  int klo = half * 2;
  v2f a0 = *(const v2f*)(ar0 + klo);
  v2f a1 = *(const v2f*)(ar1 + klo);
  v2f a2 = *(const v2f*)(ar2 + klo);
  v2f a3 = *(const v2f*)(ar3 + klo);
  v2f b  = *(const v2f*)(br  + klo);
  for (int k = 0; k < H_ - 4; k += 4) {
    const int kn = k + 4 + half * 2;
    v2f na0 = *(const v2f*)(ar0 + kn);
    v2f na1 = *(const v2f*)(ar1 + kn);
    v2f na2 = *(const v2f*)(ar2 + kn);
    v2f na3 = *(const v2f*)(ar3 + kn);
    v2f nb  = *(const v2f*)(br  + kn);
    c0 = wmma_f32(a0, b, c0);
    c1 = wmma_f32(a1, b, c1);
    c2 = wmma_f32(a2, b, c2);
    c3 = wmma_f32(a3, b, c3);
    a0 = na0; a1 = na1; a2 = na2; a3 = na3; b = nb;
  }
  c0 = wmma_f32(a0, b, c0);
  c1 = wmma_f32(a1, b, c1);
  c2 = wmma_f32(a2, b, c2);
  c3 = wmma_f32(a3, b, c3);

  const int col = tn + l15;
  const float bias = b2[o * D_ + col];
  v8f acc[4] = {c0, c1, c2, c3};
  for (int i = 0; i < 4; ++i)
    for (int v = 0; v < 8; ++v) {
      const int p = tm + 16 * i + v + half * 8;
      Z[(size_t)(p * O_ + o) * D_ + col] = acc[i][v] + bias;
    }
}

// ---------------------------------------------------------------------------
// Kernel 5: in-place row L2 normalize of Z [384 x 128]
// ---------------------------------------------------------------------------
__global__ void normalize_kernel(float* __restrict__ Z) {
  __shared__ float red[128];
  const int m = blockIdx.x;
  const int t = threadIdx.x;                // 128 threads
  const float v = Z[(size_t)m * D_ + t];
  red[t] = v * v;
  __syncthreads();
  for (int s = 64; s > 0; s >>= 1) {
    if (t < s) red[t] += red[t + s];
    __syncthreads();
  }
  const float inv = 1.0f / (sqrtf(red[0]) + EPS_);
  Z[(size_t)m * D_ + t] = v * inv;
}

// ---------------------------------------------------------------------------
// Kernel 6: S = (Zn @ Zn^T) * (1/T), 384x384. Both operands contiguous b64
// loads from Zn. 64x16 strip per wave; 144 waves -> 18 blocks * 8 waves.
// ---------------------------------------------------------------------------
__global__ void sim_kernel(const float* __restrict__ Z, float* __restrict__ S) {
  const int lane = threadIdx.x & 31;
  const int wv   = threadIdx.x >> 5;
  const int w    = blockIdx.x * 8 + wv;     // 0..143
  const int tm   = (w / 24) * 64;           // 6 strips along M=384
  const int tn   = (w % 24) * 16;           // 24 tiles along N=384
  const int half = lane >> 4;
  const int l15  = lane & 15;

  const float* ar0 = Z + (size_t)(tm +  0 + l15) * D_;
  const float* ar1 = Z + (size_t)(tm + 16 + l15) * D_;
  const float* ar2 = Z + (size_t)(tm + 32 + l15) * D_;
  const float* ar3 = Z + (size_t)(tm + 48 + l15) * D_;
  const float* br  = Z + (size_t)(tn + l15) * D_;      // B[k][n] = Zn[n][k]

  v8f c0 = {}, c1 = {}, c2 = {}, c3 = {};
  int klo = half * 2;
  v2f a0 = *(const v2f*)(ar0 + klo);
  v2f a1 = *(const v2f*)(ar1 + klo);
  v2f a2 = *(const v2f*)(ar2 + klo);
  v2f a3 = *(const v2f*)(ar3 + klo);
  v2f b  = *(const v2f*)(br  + klo);
  for (int k = 0; k < D_ - 4; k += 4) {
    const int kn = k + 4 + half * 2;
    v2f na0 = *(const v2f*)(ar0 + kn);
    v2f na1 = *(const v2f*)(ar1 + kn);
    v2f na2 = *(const v2f*)(ar2 + kn);
    v2f na3 = *(const v2f*)(ar3 + kn);
    v2f nb  = *(const v2f*)(br  + kn);
    c0 = wmma_f32(a0, b, c0);
    c1 = wmma_f32(a1, b, c1);
    c2 = wmma_f32(a2, b, c2);
    c3 = wmma_f32(a3, b, c3);
    a0 = na0; a1 = na1; a2 = na2; a3 = na3; b = nb;
  }
  c0 = wmma_f32(a0, b, c0);
  c1 = wmma_f32(a1, b, c1);
  c2 = wmma_f32(a2, b, c2);
  c3 = wmma_f32(a3, b, c3);

  const int col = tn + l15;
  v8f acc[4] = {c0, c1, c2, c3};
  for (int i = 0; i < 4; ++i)
    for (int v = 0; v < 8; ++v) {
      const int row = tm + 16 * i + v + half * 8;
      S[(size_t)row * M_ + col] = acc[i][v] * INV_T;
    }
}

// ---------------------------------------------------------------------------
// Kernel 7: per-anchor SupCon term: loss[m] = -log((pos+eps)/(denom+eps))
// ---------------------------------------------------------------------------
__global__ void rowloss_kernel(const float* __restrict__ S,
                               float* __restrict__ lossbuf) {
  __shared__ float rd[128];
  __shared__ float rp[128];
  const int m  = blockIdx.x;
  const int t  = threadIdx.x;               // 128 threads, 3 columns each
  const int pm = m / O_;
  float denom = 0.0f, pos = 0.0f;
  for (int j = t; j < M_; j += 128) {
    if (j == m) continue;                   // exp(-inf) = 0 on diagonal
    const float e = expf(S[(size_t)m * M_ + j]);
    denom += e;
    if (j / O_ == pm) pos += e;
  }
  rd[t] = denom;
  rp[t] = pos;
  __syncthreads();
  for (int s = 64; s > 0; s >>= 1) {
    if (t < s) { rd[t] += rd[t + s]; rp[t] += rp[t + s]; }
    __syncthreads();
  }
  if (t == 0) {
    lossbuf[m] = -logf((rp[0] + EPS_) / (rd[0] + EPS_));
  }
}

// ---------------------------------------------------------------------------
// Kernel 8: mean over 384 loss terms -> d_out[0]
// ---------------------------------------------------------------------------
__global__ void finalize_kernel(const float* __restrict__ lossbuf,
                                float* __restrict__ out) {
  __shared__ float red[128];
  const int t = threadIdx.x;                // 128 threads
  float s = 0.0f;
  for (int j = t; j < M_; j += 128) s += lossbuf[j];
  red[t] = s;
  __syncthreads();
  for (int k = 64; k > 0; k >>= 1) {
    if (t < k) red[t] += red[t + k];
    __syncthreads();
  }
  if (t == 0) out[0] = red[0] * (1.0f / M_);
}

// ---------------------------------------------------------------------------
extern "C" void kernel_launch(void* const* d_in, const int* in_sizes, int n_in,
                              void* d_out, int out_size, void* d_ws, size_t ws_size,
                              hipStream_t stream) {
  const float* emb = (const float*)d_in[0];   // [O,N,H]
  const float* W1  = (const float*)d_in[1];   // [O,H,H]
  const float* b1  = (const float*)d_in[2];   // [O,H]
  const float* W2  = (const float*)d_in[3];   // [O,H,D]
  const float* b2  = (const float*)d_in[4];   // [O,D]
  const int*   idx = (const int*)d_in[5];     // [O,P,G]
  float* out = (float*)d_out;

  float* ws      = (float*)d_ws;
  float* protos  = ws;                                  // O*P*H = 196608
  float* hbuf    = protos + (size_t)O_ * P_ * H_;       // O*P*H = 196608
  float* Z       = hbuf   + (size_t)O_ * P_ * H_;       // M*D   = 49152
  float* S       = Z      + (size_t)M_ * D_;            // M*M   = 147456
  float* lossbuf = S      + (size_t)M_ * M_;            // M     = 384
  float* W1t     = lossbuf + M_;                        // O*H*H = 786432
  float* W2t     = W1t + (size_t)O_ * H_ * H_;          // O*H*D = 196608

  gather_mean_kernel<<<O_ * P_, 256, 0, stream>>>(emb, idx, protos);
  transpose_kernel<<<(O_ * H_ * H_ + 255) / 256, 256, 0, stream>>>(W1, W1t, H_, H_);
  transpose_kernel<<<(O_ * H_ * D_ + 255) / 256, 256, 0, stream>>>(W2, W2t, H_, D_);
  gemm1_relu_kernel<<<24, 256, 0, stream>>>(protos, W1t, b1, hbuf);  // 192 waves
  gemm2_kernel<<<6, 256, 0, stream>>>(hbuf, W2t, b2, Z);             // 48 waves
  normalize_kernel<<<M_, 128, 0, stream>>>(Z);
  sim_kernel<<<18, 256, 0, stream>>>(Z, S);                          // 144 waves
  rowloss_kernel<<<M_, 128, 0, stream>>>(S, lossbuf);
  finalize_kernel<<<1, 128, 0, stream>>>(lossbuf, out);
}